// HoloAttention_v3_31404800868923
// MI455X (gfx1250) — compile-verified
//
#include <hip/hip_runtime.h>

#define S 384
#define H 512
#define NH 8
#define NKV 2
#define HD 64

typedef __attribute__((ext_vector_type(2))) float v2f;
typedef __attribute__((ext_vector_type(8))) float v8f;

// D = A(16x4) * B(4x16) + C, f32 in / f32 acc  (V_WMMA_F32_16X16X4_F32)
__device__ __forceinline__ v8f wmma_f32_k4(v8f acc, float ax, float ay, float bx, float by) {
  v2f a = {ax, ay};
  v2f b = {bx, by};
  return __builtin_amdgcn_wmma_f32_16x16x4_f32(false, a, false, b, (short)0, acc,
                                               false, false);
}

// ---------------- small projections: q/k/v/beta ----------------
__global__ __launch_bounds__(256) void k_proj(const float* __restrict__ signal,
                                              const float* __restrict__ beta,
                                              const float* __restrict__ qW, const float* __restrict__ qb,
                                              const float* __restrict__ kW, const float* __restrict__ kb,
                                              const float* __restrict__ vW, const float* __restrict__ vb,
                                              float* __restrict__ qsig, float* __restrict__ ksig,
                                              float* __restrict__ vsig, float* __restrict__ vbet) {
  int idx = blockIdx.x * 256 + threadIdx.x;
  if (idx >= S * 896) return;
  int s = idx / 896, c = idx % 896;
  const float* x; const float* w; const float* bb; float* out; int oc;
  if (c < 512)      { x = signal + (size_t)s * H; w = qW + (size_t)c * H;        bb = qb; oc = c;       out = qsig + (size_t)s * 512; }
  else if (c < 640) { x = signal + (size_t)s * H; w = kW + (size_t)(c - 512) * H; bb = kb; oc = c - 512; out = ksig + (size_t)s * 128; }
  else if (c < 768) { x = signal + (size_t)s * H; w = vW + (size_t)(c - 640) * H; bb = vb; oc = c - 640; out = vsig + (size_t)s * 128; }
  else              { x = beta   + (size_t)s * H; w = vW + (size_t)(c - 768) * H; bb = vb; oc = c - 768; out = vbet + (size_t)s * 128; }
  float acc = bb[oc];
  for (int h = 0; h < H; ++h) acc += x[h] * w[h];
  out[oc] = acc;
}

// ---------------- RoPE ----------------
__global__ __launch_bounds__(256) void k_rope(const float* __restrict__ qsig,
                                              const float* __restrict__ ksig,
                                              float* __restrict__ qrop, float* __restrict__ krop) {
  int idx = blockIdx.x * 256 + threadIdx.x;
  const int NQ = S * H;
  if (idx >= NQ + S * 128) return;
  // ln(10000)/32
  const float LG = 0.28782313662f;
  if (idx < NQ) {
    int s = idx >> 9, h = idx & 511, head = h >> 6, d = h & 63;
    float x = qsig[idx];
    float other = qsig[(s << 9) + (head << 6) + ((d < 32) ? d + 32 : d - 32)];
    float inv = __expf(-(float)(d & 31) * LG);
    float ang = (float)s * inv;
    qrop[head * (S * HD) + s * HD + d] = x * cosf(ang) + ((d < 32) ? -other : other) * sinf(ang);
  } else {
    int u = idx - NQ, s = u >> 7, e = u & 127, kv = e >> 6, d = e & 63;
    float x = ksig[u];
    float other = ksig[(s << 7) + (kv << 6) + ((d < 32) ? d + 32 : d - 32)];
    float inv = __expf(-(float)(d & 31) * LG);
    float ang = (float)s * inv;
    krop[kv * (S * HD) + s * HD + d] = x * cosf(ang) + ((d < 32) ? -other : other) * sinf(ang);
  }
}

// ---------------- attention scores (WMMA, K=64) + causal softmax ----------------
#define SSTR 392
__global__ __launch_bounds__(128) void k_attn(const float* __restrict__ qrop,
                                              const float* __restrict__ krop,
                                              float* __restrict__ attn) {
  __shared__ float Ss[16 * SSTR];
  __shared__ float rinv[16];
  const int head = blockIdx.y;
  const int s0 = blockIdx.x * 16;
  const int kv = head >> 2;
  const int tid = threadIdx.x, lane = tid & 31, wave = tid >> 5;
  const int row = lane & 15, sel = lane >> 4, k2 = sel * 2;
  const float* qh = qrop + (size_t)head * (S * HD);
  const float* kh = krop + (size_t)kv * (S * HD);
  for (int nt = 0; nt < 6; ++nt) {
    int j0 = (wave * 6 + nt) * 16;
    v8f acc = {0.f, 0.f, 0.f, 0.f, 0.f, 0.f, 0.f, 0.f};
    for (int k = 0; k < HD; k += 4) {
      float ax = qh[(s0 + row) * HD + k + k2];
      float ay = qh[(s0 + row) * HD + k + k2 + 1];
      float bx = kh[(j0 + row) * HD + k + k2];      // B[k, col]=K[col][k]; col==lane&15
      float by = kh[(j0 + row) * HD + k + k2 + 1];
      acc = wmma_f32_k4(acc, ax, ay, bx, by);
    }
    for (int i = 0; i < 8; ++i)
      Ss[(i + 8 * sel) * SSTR + j0 + row] = acc[i] * 0.125f;  // 1/sqrt(64)
  }
  __syncthreads();
  if (tid < 16) {
    int lim = s0 + tid + 1;                       // causal: cols 0..s inclusive
    float m = -3.4e38f;
    for (int j = 0; j < lim; ++j) m = fmaxf(m, Ss[tid * SSTR + j]);
    float sum = 0.f;
    for (int j = 0; j < lim; ++j) { float e = __expf(Ss[tid * SSTR + j] - m); Ss[tid * SSTR + j] = e; sum += e; }
    rinv[tid] = 1.f / sum;
  }
  __syncthreads();
  for (int lin = tid; lin < 16 * S; lin += 128) {
    int r = lin / S, j = lin % S;
    float v = (j <= s0 + r) ? Ss[r * SSTR + j] * rinv[r] : 0.f;
    attn[(size_t)head * (S * S) + (size_t)(s0 + r) * S + j] = v;
  }
}

// ---------------- GEMM1: v_phi[j,e,t] = sum_c vW[e,c] * phi[j,c,t] ----------------
#define BSTR 68
__global__ __launch_bounds__(128) void k_vphi(const float* __restrict__ phi,
                                              const float* __restrict__ vW,
                                              float* __restrict__ vphi) {
  __shared__ float Bs[32 * BSTR];
  const int t0 = blockIdx.x * 64;
  const int e0 = blockIdx.y * 16;
  const int j  = blockIdx.z;
  const int tid = threadIdx.x, lane = tid & 31, wave = tid >> 5;
  const int row = lane & 15, sel = lane >> 4, k2 = sel * 2;
  const float* phij = phi + (size_t)j * (H * S);
  v8f acc = {0.f, 0.f, 0.f, 0.f, 0.f, 0.f, 0.f, 0.f};
  for (int c0 = 0; c0 < H; c0 += 32) {
    __syncthreads();
    for (int it = 0; it < 16; ++it) {
      int lin = it * 128 + tid, r = lin >> 6, c = lin & 63;
      Bs[r * BSTR + c] = phij[(size_t)(c0 + r) * S + t0 + c];
    }
    __syncthreads();
    for (int kk = 0; kk < 32; kk += 4) {
      float ax = vW[(e0 + row) * H + c0 + kk + k2];
      float ay = vW[(e0 + row) * H + c0 + kk + k2 + 1];
      float bx = Bs[(kk + k2) * BSTR + wave * 16 + row];
      float by = Bs[(kk + k2 + 1) * BSTR + wave * 16 + row];
      acc = wmma_f32_k4(acc, ax, ay, bx, by);
    }
  }
  float* outp = vphi + (size_t)j * (128 * S) + (size_t)e0 * S + t0 + wave * 16;
  for (int i = 0; i < 8; ++i)
    outp[(size_t)(i + 8 * sel) * S + row] = acc[i];
}

// ------- GEMM2: out_phi[s,h,t] = sum_j attn[head,s,j]*v_phi[j,kvd,t] (causal-skipped) -------
#define ASTR 36
__global__ __launch_bounds__(128) void k_outphi(const float* __restrict__ attn,
                                                const float* __restrict__ vphi,
                                                float* __restrict__ outphi) {
  __shared__ float As[16 * ASTR];
  __shared__ float Bs[32 * BSTR];
  const int t0 = blockIdx.x * 64;
  const int s0 = blockIdx.y * 16;
  const int z  = blockIdx.z;                  // h = head*64+d
  const int head = z >> 6;
  const int e = ((head >> 2) << 6) | (z & 63);  // kv*64+d
  const int tid = threadIdx.x, lane = tid & 31, wave = tid >> 5;
  const int row = lane & 15, sel = lane >> 4, k2 = sel * 2;
  const float* arow = attn + (size_t)head * (S * S);
  const int nch = (s0 + 16 + 31) >> 5;        // attn rows are 0 beyond j>s
  v8f acc = {0.f, 0.f, 0.f, 0.f, 0.f, 0.f, 0.f, 0.f};
  for (int ch = 0; ch < nch; ++ch) {
    int j0 = ch * 32;
    __syncthreads();
    for (int it = 0; it < 4; ++it) {
      int lin = it * 128 + tid, r = lin >> 5, c = lin & 31;
      As[r * ASTR + c] = arow[(size_t)(s0 + r) * S + j0 + c];
    }
    for (int it = 0; it < 16; ++it) {
      int lin = it * 128 + tid, r = lin >> 6, c = lin & 63;
      Bs[r * BSTR + c] = vphi[(size_t)(j0 + r) * (128 * S) + (size_t)e * S + t0 + c];
    }
    __syncthreads();
    for (int kk = 0; kk < 32; kk += 4) {
      float ax = As[row * ASTR + kk + k2];
      float ay = As[row * ASTR + kk + k2 + 1];
      float bx = Bs[(kk + k2) * BSTR + wave * 16 + row];
      float by = Bs[(kk + k2 + 1) * BSTR + wave * 16 + row];
      acc = wmma_f32_k4(acc, ax, ay, bx, by);
    }
  }
  float* outp = outphi + (size_t)s0 * (H * S) + (size_t)z * S + t0 + wave * 16;
  for (int i = 0; i < 8; ++i)
    outp[(size_t)(i + 8 * sel) * (H * S) + row] = acc[i];
}

// ------- GEMM3 (in place on o_phi region): o_phi[s,o,t] = sum_h oW[o,h]*out_phi[s,h,t] -------
__global__ __launch_bounds__(256) void k_ophi(const float* __restrict__ oW,
                                              float* __restrict__ ophi) {
  __shared__ float Zs[512 * 16];              // 32 KB: full [h=512 x t=16] column block
  const int t0 = blockIdx.x * 16;
  const int s  = blockIdx.y;
  const int tid = threadIdx.x, lane = tid & 31, wave = tid >> 5;
  const int row = lane & 15, sel = lane >> 4, k2 = sel * 2;
  float* base = ophi + (size_t)s * (H * S);
  const float4* src = reinterpret_cast<const float4*>(base);
  for (int it = 0; it < 8; ++it) {
    int lin = it * 256 + tid;                 // 0..2047 float4s
    int h = lin >> 2, q = lin & 3;
    float4 v = src[(h * S + t0) / 4 + q];
    Zs[h * 16 + q * 4 + 0] = v.x; Zs[h * 16 + q * 4 + 1] = v.y;
    Zs[h * 16 + q * 4 + 2] = v.z; Zs[h * 16 + q * 4 + 3] = v.w;
  }
  __syncthreads();                            // all input staged before any in-place write
  for (int mt = 0; mt < 4; ++mt) {
    int o0 = (wave + mt * 8) * 16;
    v8f acc = {0.f, 0.f, 0.f, 0.f, 0.f, 0.f, 0.f, 0.f};
    for (int k = 0; k < H; k += 4) {
      float ax = oW[(o0 + row) * H + k + k2];
      float ay = oW[(o0 + row) * H + k + k2 + 1];
      float bx = Zs[(k + k2) * 16 + row];
      float by = Zs[(k + k2 + 1) * 16 + row];
      acc = wmma_f32_k4(acc, ax, ay, bx, by);
    }
    for (int i = 0; i < 8; ++i)
      base[(size_t)(o0 + i + 8 * sel) * S + t0 + row] = acc[i];
  }
}

// ---------------- out_sig / out_beta (attention over v_sig / v_beta) ----------------
__global__ __launch_bounds__(256) void k_outsig(const float* __restrict__ attn,
                                                const float* __restrict__ vsig,
                                                const float* __restrict__ vbet,
                                                float* __restrict__ outsig,
                                                float* __restrict__ outbet) {
  int idx = blockIdx.x * 256 + threadIdx.x;
  if (idx >= S * H) return;
  int s = idx >> 9, h = idx & 511;
  int head = h >> 6, e = ((head >> 2) << 6) | (h & 63);
  const float* ar = attn + (size_t)head * (S * S) + (size_t)s * S;
  float a1 = 0.f, a2 = 0.f;
  for (int j = 0; j <= s; ++j) {
    float a = ar[j];
    a1 += a * vsig[j * 128 + e];
    a2 += a * vbet[j * 128 + e];
  }
  outsig[idx] = a1; outbet[idx] = a2;
}

// ---------------- final o_sig / o_beta projections ----------------
__global__ __launch_bounds__(256) void k_osig(const float* __restrict__ outsig,
                                              const float* __restrict__ outbet,
                                              const float* __restrict__ oW,
                                              float* __restrict__ osig, float* __restrict__ obet) {
  int idx = blockIdx.x * 256 + threadIdx.x;
  if (idx >= S * H) return;
  int s = idx >> 9, o = idx & 511;
  const float* w = oW + (size_t)o * H;
  const float* x1 = outsig + (size_t)s * H;
  const float* x2 = outbet + (size_t)s * H;
  float a1 = 0.f, a2 = 0.f;
  for (int h = 0; h < H; ++h) { a1 += x1[h] * w[h]; a2 += x2[h] * w[h]; }
  osig[idx] = a1; obet[idx] = a2;
}

// ---------------- o_pi: top-4 |out_sig[s,h]*oW[o,h]| over h (ties -> lowest index) ----------------
__global__ __launch_bounds__(256) void k_opi(const float* __restrict__ outsig,
                                             const float* __restrict__ oW,
                                             float* __restrict__ opi) {
  int idx = blockIdx.x * 256 + threadIdx.x;
  if (idx >= S * H) return;
  int s = idx >> 9, o = idx & 511;
  const float* w = oW + (size_t)o * H;
  const float* x = outsig + (size_t)s * H;
  float v0 = -1.f, v1 = -1.f, v2 = -1.f, v3 = -1.f;
  int i0 = 0, i1 = 0, i2 = 0, i3 = 0;
  for (int h = 0; h < H; ++h) {
    float c = fabsf(x[h] * w[h]);
    if (c > v0)      { v3 = v2; i3 = i2; v2 = v1; i2 = i1; v1 = v0; i1 = i0; v0 = c; i0 = h; }
    else if (c > v1) { v3 = v2; i3 = i2; v2 = v1; i2 = i1; v1 = c; i1 = h; }
    else if (c > v2) { v3 = v2; i3 = i2; v2 = c; i2 = h; }
    else if (c > v3) { v3 = c; i3 = h; }
  }
  float* p = opi + (size_t)idx * 4;
  p[0] = (float)i0; p[1] = (float)i1; p[2] = (float)i2; p[3] = (float)i3;
}

extern "C" void kernel_launch(void* const* d_in, const int* in_sizes, int n_in,
                              void* d_out, int out_size, void* d_ws, size_t ws_size,
                              hipStream_t stream) {
  (void)in_sizes; (void)n_in; (void)out_size; (void)ws_size;
  const float* signal = (const float*)d_in[0];
  const float* phi    = (const float*)d_in[1];
  const float* beta   = (const float*)d_in[2];
  // d_in[3] = pi : unused by the reference outputs
  const float* qW = (const float*)d_in[4];
  const float* qb = (const float*)d_in[5];
  const float* kW = (const float*)d_in[6];
  const float* kb = (const float*)d_in[7];
  const float* vW = (const float*)d_in[8];
  const float* vb = (const float*)d_in[9];
  const float* oW = (const float*)d_in[10];

  float* ws = (float*)d_ws;
  float* attn   = ws;                       size_t off = (size_t)NH * S * S;
  float* vphi   = ws + off;                 off += (size_t)S * 128 * S;
  float* qsig   = ws + off;                 off += (size_t)S * H;
  float* ksig   = ws + off;                 off += (size_t)S * 128;
  float* vsig   = ws + off;                 off += (size_t)S * 128;
  float* vbet   = ws + off;                 off += (size_t)S * 128;
  float* qrop   = ws + off;                 off += (size_t)NH * S * HD;
  float* krop   = ws + off;                 off += (size_t)NKV * S * HD;
  float* outsig = ws + off;                 off += (size_t)S * H;
  float* outbet = ws + off;                 off += (size_t)S * H;

  float* out    = (float*)d_out;
  float* o_sig  = out;
  float* o_phi  = out + (size_t)S * H;                       // holds out_phi, rewritten in place
  float* o_beta = out + (size_t)S * H + (size_t)S * H * S;
  float* o_pi   = o_beta + (size_t)S * H;

  hipLaunchKernelGGL(k_proj,   dim3((S * 896 + 255) / 256), dim3(256), 0, stream,
                     signal, beta, qW, qb, kW, kb, vW, vb, qsig, ksig, vsig, vbet);
  hipLaunchKernelGGL(k_rope,   dim3((S * (H + 128) + 255) / 256), dim3(256), 0, stream,
                     qsig, ksig, qrop, krop);
  hipLaunchKernelGGL(k_attn,   dim3(S / 16, NH), dim3(128), 0, stream, qrop, krop, attn);
  hipLaunchKernelGGL(k_vphi,   dim3(S / 64, 128 / 16, S), dim3(128), 0, stream, phi, vW, vphi);
  hipLaunchKernelGGL(k_outphi, dim3(S / 64, S / 16, H), dim3(128), 0, stream, attn, vphi, o_phi);
  hipLaunchKernelGGL(k_ophi,   dim3(S / 16, S), dim3(256), 0, stream, oW, o_phi);
  hipLaunchKernelGGL(k_outsig, dim3((S * H + 255) / 256), dim3(256), 0, stream,
                     attn, vsig, vbet, outsig, outbet);
  hipLaunchKernelGGL(k_osig,   dim3((S * H + 255) / 256), dim3(256), 0, stream,
                     outsig, outbet, oW, o_sig, o_beta);
  hipLaunchKernelGGL(k_opi,    dim3((S * H + 255) / 256), dim3(256), 0, stream,
                     outsig, oW, o_pi);
}